// MaxAssigner2D_34720515620930
// MI455X (gfx1250) — compile-verified
//
#include <hip/hip_runtime.h>
#include <stdint.h>

// MaxAssigner2D: channel-max then 7-offset shift-max with zero padding.
// Bandwidth-bound (~0.28 FLOP/byte -> ~24us at 23.3 TB/s). Uses the CDNA5
// async global->LDS data mover, double-buffered on ASYNCcnt so each wave
// keeps 16 async b128 loads in flight.

#define HWDIM    512
#define CCH      32                   // channels (contiguous, 128B per pixel)
#define TDIM     64                   // output tile is TDIM x TDIM
#define HALO     2
#define MW       (TDIM + HALO)        // 66 halo tile width
#define MSTRIDE  67                   // odd stride -> conflict-free LDS
#define NPIX     (MW * MW)            // 4356 halo pixels per block
#define NTHREADS 256
#define NCHUNK   ((NPIX + NTHREADS - 1) / NTHREADS)   // 18
// dynamic LDS layout: raw double buffer (2*256*32 f32 = 64KB) + m tile
#define RAW_F32  (2 * NTHREADS * CCH)
#define SMEM_BYTES (RAW_F32 * 4 + MW * MSTRIDE * 4)

__device__ __forceinline__ float4 fmax4(float4 a, float4 b) {
    float4 r;
    r.x = fmaxf(a.x, b.x); r.y = fmaxf(a.y, b.y);
    r.z = fmaxf(a.z, b.z); r.w = fmaxf(a.w, b.w);
    return r;
}

// Issue 8 async b128 loads: 128B (32 channels) from gaddr into LDS slot.
// the instruction offset is added to both LDS and global address -> one base
// VGPR each side, offset: advances both.
__device__ __forceinline__ void async_pixel(uint32_t lds_slot,
                                            unsigned long long gaddr) {
    asm volatile(
        "global_load_async_to_lds_b128 %0, %1, off\n\t"
        "global_load_async_to_lds_b128 %0, %1, off offset:16\n\t"
        "global_load_async_to_lds_b128 %0, %1, off offset:32\n\t"
        "global_load_async_to_lds_b128 %0, %1, off offset:48\n\t"
        "global_load_async_to_lds_b128 %0, %1, off offset:64\n\t"
        "global_load_async_to_lds_b128 %0, %1, off offset:80\n\t"
        "global_load_async_to_lds_b128 %0, %1, off offset:96\n\t"
        "global_load_async_to_lds_b128 %0, %1, off offset:112"
        :
        : "v"(lds_slot), "v"(gaddr)
        : "memory");
}

__global__ __launch_bounds__(NTHREADS) void
maxassigner2d_kernel(const float* __restrict__ x, float* __restrict__ out) {
    extern __shared__ float smem[];
    float* s_raw = smem;                 // [2][NTHREADS][CCH]
    float* s_m   = smem + RAW_F32;       // [MW][MSTRIDE]

    const int tid = threadIdx.x;
    const int tx0 = blockIdx.x * TDIM;
    const int ty0 = blockIdx.y * TDIM;
    const float* xb = x + (size_t)blockIdx.z * HWDIM * HWDIM * CCH;

    // 32-bit LDS offsets of this thread's two private 128B slots
    const uint32_t slot0 = (uint32_t)(size_t)(&s_raw[tid * CCH]);
    const uint32_t slot1 = (uint32_t)(size_t)(&s_raw[(NTHREADS + tid) * CCH]);

    // global address of halo pixel for chunk k (clamped: inactive lanes do a
    // benign duplicate load into their own private slot)
    auto chunk_gaddr = [&](int k) -> unsigned long long {
        int p = k * NTHREADS + tid;
        p = (p < NPIX) ? p : (NPIX - 1);
        const int lr = p / MW;
        const int lc = p - lr * MW;
        const int r  = ty0 - HALO + lr;
        const int c  = tx0 - HALO + lc;
        const bool inb = (r >= 0) & (r < HWDIM) & (c >= 0) & (c < HWDIM);
        const int rr = inb ? r : 0;
        const int cc = inb ? c : 0;
        return (unsigned long long)(size_t)(xb + ((size_t)rr * HWDIM + cc) * CCH);
    };

    // ---- Phase 1: channel-max of the 66x66 halo tile into s_m ----
    // Software pipeline: chunk k+1 in flight while chunk k is reduced.
    async_pixel(slot0, chunk_gaddr(0));

    for (int k = 0; k < NCHUNK; ++k) {
        if (k + 1 < NCHUNK) {
            async_pixel((k & 1) ? slot0 : slot1, chunk_gaddr(k + 1));
            // 8 just-issued remain; chunk k's 8 have completed (in-order)
            asm volatile("s_wait_asynccnt 0x8" ::: "memory");
        } else {
            asm volatile("s_wait_asynccnt 0x0" ::: "memory");
        }

        const int p = k * NTHREADS + tid;
        if (p < NPIX) {
            const int lr = p / MW;
            const int lc = p - lr * MW;
            const int r  = ty0 - HALO + lr;
            const int c  = tx0 - HALO + lc;
            const bool inb = (r >= 0) & (r < HWDIM) & (c >= 0) & (c < HWDIM);

            const float4* rp = (const float4*)(&s_raw[((k & 1) ? NTHREADS : 0) * CCH + tid * CCH]);
            float4 a = fmax4(rp[0], rp[1]);
            float4 b = fmax4(rp[2], rp[3]);
            float4 d = fmax4(rp[4], rp[5]);
            float4 e = fmax4(rp[6], rp[7]);
            a = fmax4(fmax4(a, b), fmax4(d, e));
            const float m = fmaxf(fmaxf(a.x, a.y), fmaxf(a.z, a.w));

            // zero-pad semantics: OOB pixels contribute 0.0 (not -inf)
            s_m[lr * MSTRIDE + lc] = inb ? m : 0.0f;
        }
    }

    __syncthreads();

    // ---- Phase 2: 7-point shift-max over s_m ----
    // offsets (dh,dw): (0,0),(1,0),(0,1),(1,1),(2,0),(0,2),(2,2)
    for (int o = tid; o < TDIM * TDIM; o += NTHREADS) {
        const int lh = o / TDIM;
        const int lw = o - lh * TDIM;
        const int base = (lh + HALO) * MSTRIDE + (lw + HALO);

        float v = s_m[base];
        v = fmaxf(v, s_m[base - MSTRIDE]);             // (1,0)
        v = fmaxf(v, s_m[base - 1]);                   // (0,1)
        v = fmaxf(v, s_m[base - MSTRIDE - 1]);         // (1,1)
        v = fmaxf(v, s_m[base - 2 * MSTRIDE]);         // (2,0)
        v = fmaxf(v, s_m[base - 2]);                   // (0,2)
        v = fmaxf(v, s_m[base - 2 * MSTRIDE - 2]);     // (2,2)

        out[((size_t)blockIdx.z * HWDIM + (ty0 + lh)) * HWDIM + (tx0 + lw)] = v;
    }
}

extern "C" void kernel_launch(void* const* d_in, const int* in_sizes, int n_in,
                              void* d_out, int out_size, void* d_ws, size_t ws_size,
                              hipStream_t stream) {
    const float* x = (const float*)d_in[0];
    float* out = (float*)d_out;
    const int B = in_sizes[0] / (HWDIM * HWDIM * CCH);   // 16

    dim3 grid(HWDIM / TDIM, HWDIM / TDIM, B);
    maxassigner2d_kernel<<<grid, NTHREADS, SMEM_BYTES, stream>>>(x, out);
}